// SKConv_60413009985904
// MI455X (gfx1250) — compile-verified
//
#include <hip/hip_runtime.h>
#include <hip/hip_bf16.h>

// ---------------- problem constants (from reference) ----------------
#define BB   32
#define CC   128
#define HH   112
#define WW   112
#define HWN  (HH * WW)        // 12544, divisible by 4
#define DD   32
#define MM   2
#define N2   (CC * MM)        // 256
#define EPSV 1e-5f

typedef __attribute__((ext_vector_type(2))) float v2f;
typedef __attribute__((ext_vector_type(4))) float v4f;
typedef __attribute__((ext_vector_type(8))) float v8f;

// ============================================================================
// Kernel 1: bandwidth kernel.  s[b*C+c] = out1 + out2 + mean(out3[b,c,:,:])
// One workgroup per (b,c) row. 205.5 MB single-use stream > 192 MB L2, so use
// non-temporal b128 loads (TH=NT) to avoid thrashing the cache hierarchy.
// Reduction: wave32 shuffle butterfly, then one 8-entry LDS combine.
// Roofline: 205.5 MB @ 23.3 TB/s ~= 8.8 us; everything else is noise.
// ============================================================================
__global__ __launch_bounds__(256) void sk_spatial_mean(
    const float* __restrict__ out1, const float* __restrict__ out2,
    const float* __restrict__ out3, float* __restrict__ s)
{
    const int row  = blockIdx.x;                      // b*C + c, 0..4095
    const int tid  = threadIdx.x;
    const int lane = tid & 31;
    const int wave = tid >> 5;
    const v4f* p = reinterpret_cast<const v4f*>(out3 + (size_t)row * HWN);

    float acc = 0.f;
    for (int i = tid; i < HWN / 4; i += 256) {
        const v4f v = __builtin_nontemporal_load(p + i);   // global_load_b128 NT
        acc += (v.x + v.y) + (v.z + v.w);
    }

    // wave32 butterfly reduction (no LDS, no barriers)
    #pragma unroll
    for (int off = 16; off > 0; off >>= 1)
        acc += __shfl_xor(acc, off, 32);

    __shared__ float wsum[8];
    if (lane == 0) wsum[wave] = acc;
    __syncthreads();
    if (tid == 0) {
        float t = 0.f;
        #pragma unroll
        for (int w = 0; w < 8; ++w) t += wsum[w];
        s[row] = out1[row] + out2[row] + t * (1.0f / (float)HWN);
    }
}

// ============================================================================
// Kernel 2: the whole head in one 256-thread (8-wave) workgroup, fp32 WMMA.
//   y = s @ W1^T            (32x128 @ 128x32)  -> V_WMMA_F32_16X16X4_F32
//   z = relu(bn(y))
//   t = z @ W2^T            (32x32  @ 32x256)  -> V_WMMA_F32_16X16X4_F32
//   a = softmax over M=2; pooled = out1*a1 + out2*a2; out = pooled@linW^T + b
//
// WMMA operand layout (ISA 7.12.2, 32-bit 16x4 A):
//   A: lane l (l<16): {K=k0, K=k0+1}; lane l (>=16): {K=k0+2, K=k0+3}, M=l&15
//   B: same striping with N=l&15
//   C/D: VGPR r -> row (r + 8*(l>=16)), col (l&15)
// ============================================================================
__global__ __launch_bounds__(256) void sk_head(
    const float* __restrict__ out1, const float* __restrict__ out2,
    const float* __restrict__ s,                         // [32][128] (ws)
    const float* __restrict__ W1,                        // [32][128]
    const float* __restrict__ bn_gamma, const float* __restrict__ bn_beta,
    const float* __restrict__ bn_mean,  const float* __restrict__ bn_var,
    const float* __restrict__ W2,                        // [256][32]
    const float* __restrict__ linW,                      // [2][128]
    const float* __restrict__ linb,                      // [2]
    float* __restrict__ out)                             // [32][2]
{
    __shared__ float sS[BB][CC];    // staged s           (16 KB)
    __shared__ float sZ[BB][DD];    // relu(bn(y))        (4 KB)
    __shared__ float sT[BB][N2];    // z @ W2^T           (32 KB)

    const int tid  = threadIdx.x;
    const int lane = tid & 31;
    const int wave = tid >> 5;
    const int half = lane >> 4;     // 0 | 1
    const int l16  = lane & 15;

    for (int i = tid; i < BB * CC; i += 256)
        sS[i >> 7][i & 127] = s[i];
    __syncthreads();

    // ---------------- GEMM1: waves 0..3, one 16x16 tile each ----------------
    if (wave < 4) {
        const int mi = (wave >> 1) * 16;
        const int ni = (wave & 1) * 16;
        const int n  = ni + l16;                 // output column (d index)
        v8f acc = {};
        for (int k0 = 0; k0 < CC; k0 += 4) {
            const int k = k0 + 2 * half;
            v2f a; a.x = sS[mi + l16][k];  a.y = sS[mi + l16][k + 1];
            v2f b; b.x = W1[n * CC + k];   b.y = W1[n * CC + k + 1];
            acc = __builtin_amdgcn_wmma_f32_16x16x4_f32(
                false, a, false, b, (short)0, acc, false, false);
        }
        // batchnorm (running stats) + relu, scatter to LDS
        const float sc = bn_gamma[n] * rsqrtf(bn_var[n] + EPSV);
        const float sh = bn_beta[n] - bn_mean[n] * sc;
        #pragma unroll
        for (int r = 0; r < 8; ++r) {
            float zv = acc[r] * sc + sh;
            sZ[mi + half * 8 + r][n] = zv > 0.f ? zv : 0.f;
        }
    }
    __syncthreads();

    // ---------------- GEMM2: 8 waves x 4 tiles (2 x 16 tile grid) -----------
    #pragma unroll
    for (int tt = 0; tt < 4; ++tt) {
        const int tileId = wave * 4 + tt;        // 0..31
        const int mi = (tileId >> 4) * 16;       // 0 | 16
        const int ni = (tileId & 15) * 16;       // 0..240
        const int n  = ni + l16;
        v8f acc = {};
        for (int k0 = 0; k0 < DD; k0 += 4) {
            const int k = k0 + 2 * half;
            v2f a; a.x = sZ[mi + l16][k];  a.y = sZ[mi + l16][k + 1];
            v2f b; b.x = W2[n * DD + k];   b.y = W2[n * DD + k + 1];
            acc = __builtin_amdgcn_wmma_f32_16x16x4_f32(
                false, a, false, b, (short)0, acc, false, false);
        }
        #pragma unroll
        for (int r = 0; r < 8; ++r)
            sT[mi + half * 8 + r][n] = acc[r];
    }
    __syncthreads();

    // ------------- softmax(M=2) + pooled + final linear ---------------------
    if (tid < BB * 2) {
        const int b = tid >> 1;
        const int i = tid & 1;
        float acc = 0.f;
        for (int c = 0; c < CC; ++c) {
            const float t1 = sT[b][c], t2 = sT[b][CC + c];
            const float mx = fmaxf(t1, t2);
            const float e1 = __expf(t1 - mx), e2 = __expf(t2 - mx);
            const float a1 = e1 / (e1 + e2);
            const float pooled = out1[b * CC + c] * a1 + out2[b * CC + c] * (1.f - a1);
            acc += pooled * linW[i * CC + c];
        }
        out[b * 2 + i] = acc + linb[i];
    }
}

// ============================================================================
extern "C" void kernel_launch(void* const* d_in, const int* in_sizes, int n_in,
                              void* d_out, int out_size, void* d_ws, size_t ws_size,
                              hipStream_t stream) {
    const float* out1     = (const float*)d_in[0];   // (32,128)
    const float* out2     = (const float*)d_in[1];   // (32,128)
    const float* out3     = (const float*)d_in[2];   // (32,128,112,112)
    const float* W1       = (const float*)d_in[3];   // (32,128)
    const float* bn_gamma = (const float*)d_in[4];   // (32,)
    const float* bn_beta  = (const float*)d_in[5];   // (32,)
    const float* bn_mean  = (const float*)d_in[6];   // (32,)
    const float* bn_var   = (const float*)d_in[7];   // (32,)
    const float* W2       = (const float*)d_in[8];   // (256,32)
    const float* linW     = (const float*)d_in[9];   // (2,128)
    const float* linb     = (const float*)d_in[10];  // (2,)
    float* out = (float*)d_out;                      // (32,2)

    float* s = (float*)d_ws;                         // 4096 floats scratch

    sk_spatial_mean<<<BB * CC, 256, 0, stream>>>(out1, out2, out3, s);
    sk_head<<<1, 256, 0, stream>>>(out1, out2, s, W1, bn_gamma, bn_beta,
                                   bn_mean, bn_var, W2, linW, linb, out);
}